// QuantumPulseGaussian_20194936226273
// MI455X (gfx1250) — compile-verified
//
#include <hip/hip_runtime.h>
#include <hip/hip_bf16.h>

#define DIMN   128
#define PLANE  (DIMN * DIMN)     // 16384 floats per real/imag plane
#define NSQ    14                // fixed squaring count (2^-14 scaling)
#define NTERM  8                 // Taylor order: I + X + ... + X^8/8!

typedef __attribute__((ext_vector_type(2))) float v2f;
typedef __attribute__((ext_vector_type(8))) float v8f;

// ---------------------------------------------------------------------------
// Kernel 1: pulse area  a = dt*dx*mag * sum_t exp(-(x_t-mu)^2 / (2 sigma^2))
// writes coef = a * 2^-NSQ
// ---------------------------------------------------------------------------
__global__ void pulse_coef_kernel(const float* __restrict__ pp,
                                  float* __restrict__ coef) {
    __shared__ float red[256];
    const float mag = pp[0], mu = pp[1], sigma = pp[2];
    const float inv2s2 = 1.0f / (2.0f * sigma * sigma);
    float sum = 0.0f;
    for (int t = (int)threadIdx.x; t < 2000; t += 256) {
        float x = -10.0f + 0.01f * (float)t;
        float d = x - mu;
        sum += expf(-d * d * inv2s2);
    }
    red[threadIdx.x] = sum;
    __syncthreads();
    for (int s = 128; s > 0; s >>= 1) {
        if ((int)threadIdx.x < s) red[threadIdx.x] += red[threadIdx.x + s];
        __syncthreads();
    }
    if (threadIdx.x == 0) {
        float a = mag * 1.0f /*dt*/ * 0.01f /*dx*/ * red[0];
        coef[0] = a * (1.0f / (float)(1 << NSQ));
    }
}

// ---------------------------------------------------------------------------
// Kernel 2: X = (-i * coef) * H = coef*Hi - i*coef*Hr ; S = I + X ; T = X
// ---------------------------------------------------------------------------
__global__ void build_kernel(const float* __restrict__ Hr, const float* __restrict__ Hi,
                             const float* __restrict__ coef,
                             float* __restrict__ Xr, float* __restrict__ Xi,
                             float* __restrict__ Sr, float* __restrict__ Si,
                             float* __restrict__ Tr, float* __restrict__ Ti) {
    const int idx = blockIdx.x * blockDim.x + threadIdx.x;
    const float c  = coef[0];
    const float xr =  c * Hi[idx];
    const float xi = -c * Hr[idx];
    Xr[idx] = xr;  Xi[idx] = xi;
    const int row = idx >> 7, col = idx & (DIMN - 1);
    const float eye = (row == col) ? 1.0f : 0.0f;
    Sr[idx] = eye + xr;  Si[idx] = xi;
    Tr[idx] = xr;        Ti[idx] = xi;
}

// ---------------------------------------------------------------------------
// Complex 128x128 GEMM on fp32 WMMA:  C = scale * (A @ B)
//   optional: acc += C (Taylor-term accumulation into S)
//   optional: write interleaved complex64 to outI (final result)
// One wave per 16x16 output tile; 64 waves total (16 blocks x 128 threads).
// K-loop fully unrolled: 32 steps x 4 v_wmma_f32_16x16x4_f32.
// ---------------------------------------------------------------------------
__global__ void zgemm128_kernel(const float* __restrict__ Ar, const float* __restrict__ Ai,
                                const float* __restrict__ Br, const float* __restrict__ Bi,
                                float* __restrict__ Cr, float* __restrict__ Ci,
                                float* __restrict__ accR, float* __restrict__ accI,
                                float scale, float* __restrict__ outI) {
    const int lane = threadIdx.x & 31;
    const int wave = threadIdx.x >> 5;
    const int tile = blockIdx.x * 4 + wave;     // 0..63
    const int tm = tile >> 3;                   // tile row
    const int tn = tile & 7;                    // tile col
    const int m    = lane & 15;                 // A: M index, B/C: N index
    const int khi  = (lane >> 4) << 1;          // lanes 16..31 hold K+2,K+3
    const int rowA = (tm * 16 + m) * DIMN;
    const int colB = tn * 16 + m;

    v8f cr = {};
    v8f ci = {};
#pragma unroll
    for (int kk = 0; kk < DIMN; kk += 4) {
        const int ka = kk + khi;                // even -> 8B aligned pair load
        v2f ar = *(const v2f*)(Ar + rowA + ka);
        v2f ai = *(const v2f*)(Ai + rowA + ka);
        v2f ani; ani.x = -ai.x; ani.y = -ai.y;  // f32 WMMA has no A-neg: negate in VALU
        v2f br;  br.x = Br[ka * DIMN + colB];  br.y = Br[(ka + 1) * DIMN + colB];
        v2f bi;  bi.x = Bi[ka * DIMN + colB];  bi.y = Bi[(ka + 1) * DIMN + colB];
        // Cr += Ar*Br - Ai*Bi ; Ci += Ar*Bi + Ai*Br
        cr = __builtin_amdgcn_wmma_f32_16x16x4_f32(false, ar,  false, br, (short)0, cr, false, false);
        cr = __builtin_amdgcn_wmma_f32_16x16x4_f32(false, ani, false, bi, (short)0, cr, false, false);
        ci = __builtin_amdgcn_wmma_f32_16x16x4_f32(false, ar,  false, bi, (short)0, ci, false, false);
        ci = __builtin_amdgcn_wmma_f32_16x16x4_f32(false, ai,  false, br, (short)0, ci, false, false);
    }

    // C/D layout: VGPR r, lanes 0-15 -> M=r, lanes 16-31 -> M=r+8; N=lane&15
    const int rbase = tm * 16 + ((lane >> 4) << 3);
    const int col   = tn * 16 + m;
#pragma unroll
    for (int r = 0; r < 8; r++) {
        const float vr = cr[r] * scale;
        const float vi = ci[r] * scale;
        const int idx = (rbase + r) * DIMN + col;
        if (accR) { accR[idx] += vr; accI[idx] += vi; }   // S += X^k/k!
        if (Cr)   { Cr[idx] = vr;   Ci[idx] = vi; }       // term / square planes
        if (outI) { outI[2 * idx] = vr; outI[2 * idx + 1] = vi; } // complex64 out
    }
}

// ---------------------------------------------------------------------------
extern "C" void kernel_launch(void* const* d_in, const int* in_sizes, int n_in,
                              void* d_out, int out_size, void* d_ws, size_t ws_size,
                              hipStream_t stream) {
    const float* Hr = (const float*)d_in[0];   // hamil_real  [128,128]
    const float* Hi = (const float*)d_in[1];   // hamil_imag  [128,128]
    const float* pp = (const float*)d_in[2];   // pulse_params [3]
    float* out = (float*)d_out;                // complex64 -> interleaved f32 pairs

    // Workspace layout (floats): coef pad(64) | X(2P) | S(2P) | T0(2P) | T1(2P)
    float* w    = (float*)d_ws;
    float* coef = w;
    float* Xr   = w + 64;          float* Xi  = Xr + PLANE;
    float* Sr   = Xi + PLANE;      float* Si  = Sr + PLANE;
    float* T0r  = Si + PLANE;      float* T0i = T0r + PLANE;
    float* T1r  = T0i + PLANE;     float* T1i = T1r + PLANE;

    // 1) pulse area -> scaled coefficient
    pulse_coef_kernel<<<1, 256, 0, stream>>>(pp, coef);

    // 2) X = (-i*coef)*H ; S = I + X ; T0 = X
    build_kernel<<<PLANE / 256, 256, 0, stream>>>(Hr, Hi, coef, Xr, Xi, Sr, Si, T0r, T0i);

    // 3) Taylor terms: T_k = T_{k-1} @ X / k ; S += T_k   (k = 2..NTERM)
    float* tinr = T0r; float* tini = T0i;
    float* totr = T1r; float* toti = T1i;
    for (int k = 2; k <= NTERM; ++k) {
        zgemm128_kernel<<<16, 128, 0, stream>>>(tinr, tini, Xr, Xi,
                                                totr, toti, Sr, Si,
                                                1.0f / (float)k, nullptr);
        float* tr = tinr; tinr = totr; totr = tr;
        float* ti = tini; tini = toti; toti = ti;
    }

    // 4) Squaring chain: U <- U@U, NSQ times. Input q==0 is S; outputs ping-pong
    //    T0/T1; last squaring writes interleaved complex64 directly to d_out.
    const float* ur = Sr; const float* ui = Si;
    for (int q = 0; q < NSQ; ++q) {
        const bool last = (q == NSQ - 1);
        float* or_ = (q & 1) ? T1r : T0r;
        float* oi_ = (q & 1) ? T1i : T0i;
        zgemm128_kernel<<<16, 128, 0, stream>>>(ur, ui, ur, ui,
                                                last ? nullptr : or_,
                                                last ? nullptr : oi_,
                                                nullptr, nullptr,
                                                1.0f, last ? out : nullptr);
        ur = or_; ui = oi_;
    }
}